// LengthRegulator_55052890800577
// MI455X (gfx1250) — compile-verified
//
#include <hip/hip_runtime.h>

// ---------------------------------------------------------------------------
// LengthRegulator for MI455X (gfx1250).
// Memory-bound row gather/expand: ~256MB stores + ~64MB gathered loads.
// Uses CDNA5 async LDS DMA (GLOBAL_LOAD_ASYNC_TO_LDS_B128 /
// GLOBAL_STORE_ASYNC_FROM_LDS_B128, ASYNCcnt) for the row copies so the
// payload never round-trips through VGPRs.
// ---------------------------------------------------------------------------

#define AS1 __attribute__((address_space(1)))
#define AS3 __attribute__((address_space(3)))

// The gfx1250 b128 async builtins take pointers to a 16-byte int vector.
typedef int v4i __attribute__((vector_size(16)));

#if defined(__gfx1250__) && \
    __has_builtin(__builtin_amdgcn_global_load_async_to_lds_b128) && \
    __has_builtin(__builtin_amdgcn_global_store_async_from_lds_b128)
#define USE_ASYNC 1
#else
#define USE_ASYNC 0
#endif

__device__ __forceinline__ void wait_async0() {
#if defined(__gfx1250__)
#if __has_builtin(__builtin_amdgcn_s_wait_asynccnt)
  __builtin_amdgcn_s_wait_asynccnt(0);
#else
  asm volatile("s_wait_asynccnt 0x0" ::: "memory");
#endif
#endif
}

#if USE_ASYNC
template <int OFF>
__device__ __forceinline__ void async_g2l_b128(const float* g, float* l) {
  // INST_OFFSET applies to both the global and the LDS address (ISA 08 §4.4),
  // so one base pointer pair serves all 4 chunks of a 2KB row.
  __builtin_amdgcn_global_load_async_to_lds_b128((AS1 v4i*)(g), (AS3 v4i*)(l), OFF, 0);
}
template <int OFF>
__device__ __forceinline__ void async_l2g_b128(float* g, const float* l) {
  __builtin_amdgcn_global_store_async_from_lds_b128((AS1 v4i*)(g), (AS3 v4i*)(l), OFF, 0);
}
#endif

// ---------------------------------------------------------------------------
// Kernel 1: per-batch inclusive cumsum of durations -> cum (int32, B*S in ws),
// and output_lengths (float) appended at the tail of d_out.
// ---------------------------------------------------------------------------
__global__ __launch_bounds__(256) void lr_cumsum_kernel(
    const int* __restrict__ dur, int* __restrict__ cum,
    float* __restrict__ out_len, int S) {
  __shared__ int part[256];
  const int b = blockIdx.x;
  const int tid = threadIdx.x;
  const int* drow = dur + (size_t)b * S;

  int per = (S + 255) >> 8;      // elements per thread (4 for S=1024)
  if (per > 8) per = 8;
  const int base = tid * per;

  int local[8];
  int s = 0;
  for (int i = 0; i < per; ++i) {
    int v = (base + i < S) ? drow[base + i] : 0;
    local[i] = v;
    s += v;
  }
  part[tid] = s;
  __syncthreads();

  // Hillis-Steele inclusive scan over 256 partials.
  for (int off = 1; off < 256; off <<= 1) {
    int add = (tid >= off) ? part[tid - off] : 0;
    __syncthreads();
    part[tid] += add;
    __syncthreads();
  }

  int run = part[tid] - s;       // exclusive prefix of this thread's chunk
  for (int i = 0; i < per; ++i) {
    run += local[i];
    if (base + i < S) cum[(size_t)b * S + base + i] = run;
  }
  if (tid == 255) out_len[b] = (float)part[255];
}

// ---------------------------------------------------------------------------
// Kernel 2: expand. One block = 8 waves = one 64-frame tile of one batch.
// Each wave copies 8 consecutive 2KB rows via async LDS DMA.
// Assumes H == 512 on the async path (compile-time 4x b128 per lane).
// ---------------------------------------------------------------------------
#define TILE_T 64

__global__ __launch_bounds__(256) void lr_expand_kernel(
    const float* __restrict__ x, const int* __restrict__ cum,
    float* __restrict__ out, int S, int H, int max_len) {
  __shared__ int   scum[1024];       // cum row (S <= 1024)
  __shared__ float zbuf[512];        // zero row for padded frames
  __shared__ float stage[8][512];    // 2KB staging buffer per wave

  const int tid  = threadIdx.x;
  const int wave = tid >> 5;
  const int lane = tid & 31;

  const int tiles_per_b = (max_len + TILE_T - 1) / TILE_T;
  const int b  = blockIdx.x / tiles_per_b;
  const int t0 = (blockIdx.x % tiles_per_b) * TILE_T;

  const int* crow = cum + (size_t)b * S;

  // Stage the cum row into LDS (async b128: 16B per thread covers S*4 bytes).
#if USE_ASYNC
  {
    const int nvec = S >> 2;
    for (int i = tid; i < nvec; i += 256)
      async_g2l_b128<0>((const float*)(crow + i * 4), (float*)&scum[i * 4]);
    wait_async0();
  }
#else
  for (int i = tid; i < S; i += 256) scum[i] = crow[i];
#endif
  for (int i = tid; i < 512; i += 256) zbuf[i] = 0.0f;
  __syncthreads();

  const int len = scum[S - 1];
  const bool fast = (H == 512);

  for (int j = 0; j < TILE_T / 8; ++j) {
    const int t = t0 + wave * (TILE_T / 8) + j;
    if (t >= max_len) break;
    float* dst = out + ((size_t)b * max_len + t) * (size_t)H;

    if (t < len) {
      // searchsorted(cum, t, 'right'): first idx with scum[idx] > t.
      int lo = 0, hi = S;
      while (lo < hi) {
        int mid = (lo + hi) >> 1;
        if (scum[mid] <= t) lo = mid + 1; else hi = mid;
      }
      const int idx = (lo < S) ? lo : (S - 1);
      const float* src = x + ((size_t)b * S + idx) * (size_t)H;

#if USE_ASYNC
      if (fast) {
        float*       st = &stage[wave][lane * 4];
        const float* gs = src + lane * 4;
        wait_async0();                 // prior stores from stage[wave] done
        async_g2l_b128<0>(gs, st);
        async_g2l_b128<512>(gs, st);
        async_g2l_b128<1024>(gs, st);
        async_g2l_b128<1536>(gs, st);
        wait_async0();                 // stage filled
        float* gd = dst + lane * 4;
        async_l2g_b128<0>(gd, st);
        async_l2g_b128<512>(gd, st);
        async_l2g_b128<1024>(gd, st);
        async_l2g_b128<1536>(gd, st);
        continue;
      }
#endif
      for (int k = lane; k < (H >> 2); k += 32)
        ((float4*)dst)[k] = ((const float4*)src)[k];
    } else {
#if USE_ASYNC
      if (fast) {
        const float* zl = &zbuf[lane * 4];
        float*       gd = dst + lane * 4;
        async_l2g_b128<0>(gd, zl);
        async_l2g_b128<512>(gd, zl);
        async_l2g_b128<1024>(gd, zl);
        async_l2g_b128<1536>(gd, zl);
        continue;
      }
#endif
      const float4 z = make_float4(0.f, 0.f, 0.f, 0.f);
      for (int k = lane; k < (H >> 2); k += 32) ((float4*)dst)[k] = z;
    }
  }
  wait_async0();   // drain before S_ENDPGM (which also implies wait-idle)
}

// ---------------------------------------------------------------------------
// Launch: inputs are [x (B*S*H f32), durations (B*S int32), max_len (1 int)].
// d_out = expanded (B*max_len*H f32) ++ output_lengths (B).
// ---------------------------------------------------------------------------
extern "C" void kernel_launch(void* const* d_in, const int* in_sizes, int n_in,
                              void* d_out, int out_size, void* d_ws, size_t ws_size,
                              hipStream_t stream) {
  const float* x   = (const float*)d_in[0];
  const int*   dur = (const int*)d_in[1];

  const int BS = in_sizes[1];           // B*S
  const int H  = in_sizes[0] / BS;      // 512
  const int B  = 32;                    // fixed problem shape
  const int S  = BS / B;                // 1024
  const int max_len = (int)(((long long)out_size - B) / ((long long)B * H)); // 4096

  float* out     = (float*)d_out;
  float* out_len = out + (size_t)B * max_len * H;
  int*   cum     = (int*)d_ws;          // B*S int32 scratch (128 KB)

  lr_cumsum_kernel<<<B, 256, 0, stream>>>(dur, cum, out_len, S);

  const int tiles_per_b = (max_len + TILE_T - 1) / TILE_T;
  lr_expand_kernel<<<B * tiles_per_b, 256, 0, stream>>>(x, cum, out, S, H, max_len);
}